// TransE_69312182222861
// MI455X (gfx1250) — compile-verified
//
#include <hip/hip_runtime.h>
#include <stdint.h>

// ---------------------------------------------------------------------------
// TransE scoring on MI455X (gfx1250, wave32, WGP).
//   scores_sp[b,e] = -sum_k |(lhs[b]+rel[b])[k] - W[e,k]|
//   scores_po[b,e] = -sum_k |W[e,k] - (rhs[b]-rel[b])[k]|
// W (40000x256 f32) is streamed through double-buffered LDS tiles with the
// CDNA5 Tensor Data Mover (inline-asm `tensor_load_to_lds`, TENSORcnt waits),
// q/r per-batch vectors live in LDS, threads register-block 2x2 (batch x
// entity), accumulation is v_sub / v_add-with-|src| VALU work (an L1 norm has
// no matmul structure, so WMMA does not apply here).
// ---------------------------------------------------------------------------

#define NUM_ENT 40000
#define RANK    256
#define BATCH   64

#define ETS  160           // entities per workgroup (supertile)
#define ESUB 32            // entities per LDS subtile
#define NSUB (ETS / ESUB)  // 5 subtiles, ping-pong across 2 buffers
#define BH   32            // batches per workgroup (batch split in 2)
#define ROWP 260           // padded LDS row stride in floats (256 + 4)

// Output layout (flat, in reference return order)
#define SP_OFF  0
#define PO_OFF  (BATCH * NUM_ENT)
#define LHS_OFF (2 * BATCH * NUM_ENT)
#define REL_OFF (LHS_OFF + BATCH * RANK)
#define RHS_OFF (REL_OFF + BATCH * RANK)

// Reference dtype of x is int64, but harness doc maps integer inputs to int32.
#ifndef X_INT64
#define X_INT64 0
#endif
#if X_INT64
typedef long long xint_t;
#else
typedef int xint_t;
#endif

// Flip to 1 if the assembler requires the explicit 4-group operand form.
#ifndef TDM_FOUR_GROUPS
#define TDM_FOUR_GROUPS 0
#endif

typedef __attribute__((ext_vector_type(4))) unsigned tdm_v4u;
typedef __attribute__((ext_vector_type(8))) unsigned tdm_v8u;

// Issue one 32x256-f32 tile load (global -> LDS) via the Tensor Data Mover.
// The D# enables LDS padding (pad_interval = 256 DW, pad_amount = 4 DW) so
// rows land at a bank-friendly stride of ROWP floats.
__device__ __forceinline__ void tile_load(float* lds_dst, const float* gsrc,
                                          int rows_remaining, int tid) {
#if defined(__gfx1250__)
  if (tid == 0) {
    unsigned long long ga = (unsigned long long)(uintptr_t)gsrc;
    unsigned lds_off = (unsigned)(uintptr_t)lds_dst;  // low 32 bits == LDS offset

    tdm_v4u g0;
    g0.x = 1u;                                              // count=1, user mode
    g0.y = lds_off;                                         // lds_addr (bytes)
    g0.z = (unsigned)ga;                                    // global_addr[31:0]
    g0.w = (unsigned)((ga >> 32) & 0x01FFFFFFull)           // global_addr[56:32]
           | (2u << 30);                                    // type=2 ("image")

    tdm_v8u g1;
    // data_size=4B (2<<16), pad_enable(bit20), pad_interval=7 (256 DW),
    // pad_amount=3 (4 DW). workgroup_mask=0 (not in cluster).
    g1[0] = 0x00020000u | (1u << 20) | (7u << 22) | (3u << 25);
    g1[1] = ((unsigned)RANK & 0xFFFFu) << 16;               // tensor_dim0 lo16
    g1[2] = (((unsigned)RANK >> 16) & 0xFFFFu)              // tensor_dim0 hi16
          | (((unsigned)rows_remaining & 0xFFFFu) << 16);   // tensor_dim1 lo16
    g1[3] = (((unsigned)rows_remaining >> 16) & 0xFFFFu)    // tensor_dim1 hi16
          | (((unsigned)RANK & 0xFFFFu) << 16);             // tile_dim0 = 256
    g1[4] = (unsigned)ESUB;                                 // tile_dim1=32, tile_dim2=0
    g1[5] = (unsigned)RANK;                                 // tensor_dim0_stride lo32
    g1[6] = 0u;                                             // stride hi / dim1_stride lo
    g1[7] = 0u;

#if TDM_FOUR_GROUPS
    tdm_v4u z0 = {0u, 0u, 0u, 0u};
    tdm_v4u z1 = {0u, 0u, 0u, 0u};
    asm volatile("tensor_load_to_lds %0, %1, %2, %3"
                 :: "s"(g0), "s"(g1), "s"(z0), "s"(z1)
                 : "memory");
#else
    // VADDR2/VADDR3 omitted -> NULL (2D tensor form).
    asm volatile("tensor_load_to_lds %0, %1"
                 :: "s"(g0), "s"(g1)
                 : "memory");
#endif
  }
#else
  // Non-gfx1250 / host parse path: cooperative copy with the same LDS layout.
  (void)rows_remaining;
  for (int idx = tid; idx < ESUB * (RANK / 4); idx += 256) {
    int row = idx >> 6;
    int c   = (idx & 63) << 2;
    *(float4*)(lds_dst + row * ROWP + c) = *(const float4*)(gsrc + row * RANK + c);
  }
#endif
}

__device__ __forceinline__ void tile_wait(int tid) {
#if defined(__gfx1250__)
  if (tid == 0) {
#if __has_builtin(__builtin_amdgcn_s_wait_tensorcnt)
    __builtin_amdgcn_s_wait_tensorcnt(0);
#else
    asm volatile("s_wait_tensorcnt 0x0" ::: "memory");
#endif
  }
#else
  (void)tid;
#endif
}

__device__ __forceinline__ float l1_4(float4 a, float4 b) {
  return fabsf(a.x - b.x) + fabsf(a.y - b.y) + fabsf(a.z - b.z) + fabsf(a.w - b.w);
}

__global__ __launch_bounds__(256) void
transe_score_kernel(const xint_t* __restrict__ x,
                    const float* __restrict__ lhs_w,
                    const float* __restrict__ rel_w,
                    float* __restrict__ out) {
  __shared__ __align__(16) float sW[2][ESUB * ROWP];  // 2 x 33.3 KB ping-pong
  __shared__ __align__(16) float sQ[BH * ROWP];       // q = lhs + rel
  __shared__ __align__(16) float sR[BH * ROWP];       // r = rhs - rel

  const int tid   = threadIdx.x;
  const int row0  = blockIdx.x * ETS;   // first entity of this supertile
  const int bbase = blockIdx.y * BH;    // first batch of this half

  // Kick off the first async tile while we build q/r.
  tile_load(&sW[0][0], lhs_w + (size_t)row0 * RANK, NUM_ENT - row0, tid);

  // Build q/r tiles in LDS (gathers are tiny; L2-hot after first touch).
  for (int idx = tid; idx < BH * RANK; idx += 256) {
    int bl = idx >> 8;          // RANK == 256
    int k  = idx & (RANK - 1);
    int b  = bbase + bl;
    int i0 = (int)x[b * 3 + 0];
    int i1 = (int)x[b * 3 + 1];
    int i2 = (int)x[b * 3 + 2];
    float lv = lhs_w[(size_t)i0 * RANK + k];
    float rv = rel_w[(size_t)i1 * RANK + k];
    float hv = lhs_w[(size_t)i2 * RANK + k];
    sQ[bl * ROWP + k] = lv + rv;
    sR[bl * ROWP + k] = hv - rv;
  }

  // Auxiliary outputs (lhs, rel, rhs gathers) from a single block.
  if (blockIdx.x == 0 && blockIdx.y == 0) {
    for (int idx = tid; idx < BATCH * RANK; idx += 256) {
      int b = idx >> 8;
      int k = idx & (RANK - 1);
      int i0 = (int)x[b * 3 + 0];
      int i1 = (int)x[b * 3 + 1];
      int i2 = (int)x[b * 3 + 2];
      out[LHS_OFF + idx] = lhs_w[(size_t)i0 * RANK + k];
      out[REL_OFF + idx] = rel_w[(size_t)i1 * RANK + k];
      out[RHS_OFF + idx] = lhs_w[(size_t)i2 * RANK + k];
    }
  }

  // Thread tiling: 16 entity-groups x 16 batch-groups; 2 entities x 2 batches
  // per thread. Within a wave, 16 lanes span entities (ROWP stride -> at most
  // 2-way LDS bank conflict on b128 reads) and q/r reads are broadcasts.
  const int te = tid & 15;
  const int tb = tid >> 4;
  const int e0 = te * 2;
  const int b0 = tb * 2;
  const float* qp0 = &sQ[(b0 + 0) * ROWP];
  const float* qp1 = &sQ[(b0 + 1) * ROWP];
  const float* rp0 = &sR[(b0 + 0) * ROWP];
  const float* rp1 = &sR[(b0 + 1) * ROWP];

  for (int s = 0; s < NSUB; ++s) {
    tile_wait(tid);       // TDM for subtile s complete (wave 0)
    __syncthreads();      // tile s visible to all; previous buffer free
    if (s + 1 < NSUB) {
      int nrow = row0 + (s + 1) * ESUB;
      tile_load(&sW[(s + 1) & 1][0], lhs_w + (size_t)nrow * RANK,
                NUM_ENT - nrow, tid);   // prefetch next subtile async
    }

    const float* w0p = &sW[s & 1][(e0 + 0) * ROWP];
    const float* w1p = &sW[s & 1][(e0 + 1) * ROWP];

    float sp00 = 0.f, sp01 = 0.f, sp10 = 0.f, sp11 = 0.f;
    float po00 = 0.f, po01 = 0.f, po10 = 0.f, po11 = 0.f;

#pragma unroll 4
    for (int k = 0; k < RANK; k += 4) {
      float4 w0 = *(const float4*)(w0p + k);
      float4 w1 = *(const float4*)(w1p + k);
      float4 q0 = *(const float4*)(qp0 + k);
      float4 q1 = *(const float4*)(qp1 + k);
      float4 r0 = *(const float4*)(rp0 + k);
      float4 r1 = *(const float4*)(rp1 + k);
      sp00 += l1_4(q0, w0);  sp01 += l1_4(q0, w1);
      sp10 += l1_4(q1, w0);  sp11 += l1_4(q1, w1);
      po00 += l1_4(w0, r0);  po01 += l1_4(w1, r0);
      po10 += l1_4(w0, r1);  po11 += l1_4(w1, r1);
    }

    const int    eg  = row0 + s * ESUB + e0;
    const size_t rb0 = (size_t)(bbase + b0 + 0) * NUM_ENT;
    const size_t rb1 = (size_t)(bbase + b0 + 1) * NUM_ENT;
    out[SP_OFF + rb0 + eg + 0] = -sp00;
    out[SP_OFF + rb0 + eg + 1] = -sp01;
    out[SP_OFF + rb1 + eg + 0] = -sp10;
    out[SP_OFF + rb1 + eg + 1] = -sp11;
    out[PO_OFF + rb0 + eg + 0] = -po00;
    out[PO_OFF + rb0 + eg + 1] = -po01;
    out[PO_OFF + rb1 + eg + 0] = -po10;
    out[PO_OFF + rb1 + eg + 1] = -po11;
  }
}

extern "C" void kernel_launch(void* const* d_in, const int* in_sizes, int n_in,
                              void* d_out, int out_size, void* d_ws, size_t ws_size,
                              hipStream_t stream) {
  (void)in_sizes; (void)n_in; (void)out_size; (void)d_ws; (void)ws_size;
  const xint_t* x   = (const xint_t*)d_in[0];
  const float*  lhs = (const float*)d_in[1];
  const float*  rel = (const float*)d_in[2];
  float*        out = (float*)d_out;

  dim3 grid(NUM_ENT / ETS, BATCH / BH, 1);  // 250 x 2 workgroups
  transe_score_kernel<<<grid, 256, 0, stream>>>(x, lhs, rel, out);
}